// TransitionStructure_36790689857547
// MI455X (gfx1250) — compile-verified
//
#include <hip/hip_runtime.h>

// CDNA5 / gfx1250, wave32. Memory-bound masked matvec:
//   raw[e]   = rel_emb[e,:64] . W[512:576] + (q . W[:512] + b)
//   score[e] = match ? raw*conf*weight : 0 ;  tgt[e] = match ? target : -1
// Dot products run on V_WMMA_F32_16X16X4_F32 (fp32-exact matrix path).

typedef __attribute__((ext_vector_type(2))) float v2f;
typedef __attribute__((ext_vector_type(8))) float v8f;

#define D_REL 64

// ---------------- kernel 1: q_score + b -> ws[0] ----------------
__global__ __launch_bounds__(512)
void qscore_kernel(const float* __restrict__ q, const float* __restrict__ W,
                   const float* __restrict__ b, float* __restrict__ out,
                   int d_model) {
    __shared__ float red[16];
    int t = threadIdx.x;
    float v = (t < d_model) ? q[t] * W[t] : 0.0f;
    #pragma unroll
    for (int off = 16; off > 0; off >>= 1)
        v += __shfl_down(v, off, 32);
    if ((t & 31) == 0) red[t >> 5] = v;
    __syncthreads();
    if (t < 16) {
        float s = red[t];
        #pragma unroll
        for (int off = 8; off > 0; off >>= 1)
            s += __shfl_down(s, off, 16);
        if (t == 0) out[0] = s + b[0];
    }
}

// ---------------- kernel 2: per-edge scores via WMMA ----------------
// One wave32 handles a 16-edge tile. A (16x4 fp32): lanes 0-15 hold K=k0,k0+1,
// lanes 16-31 hold K=k0+2,k0+3 (ISA 7.12.2). B (4x16): W chunk broadcast over
// all N columns -> every column of D is the per-row dot product.
__global__ __launch_bounds__(256)
void transition_kernel(const float* __restrict__ rel,     // E x 64
                       const float* __restrict__ Wr,      // W + 512
                       const float* __restrict__ weight,
                       const float* __restrict__ conf,
                       const int*   __restrict__ source,
                       const int*   __restrict__ target,
                       const unsigned char* __restrict__ mask,
                       const float* __restrict__ qsb,     // scalar q.W + b
                       const int*   __restrict__ basin_p,
                       float* __restrict__ scores,
                       float* __restrict__ tgt_out,
                       int E) {
    const int lane = threadIdx.x & 31;
    const int wave = threadIdx.x >> 5;
    const int tile = blockIdx.x * 8 + wave;          // 16 edges per tile
    const int numFull  = E >> 4;
    const int numTiles = (E + 15) >> 4;
    if (tile >= numTiles) return;                    // wave-uniform
    const float qs    = qsb[0];
    const int   basin = basin_p[0];

    if (tile < numFull) {
        // -------- WMMA path: EXEC is all-ones here (wave-uniform branch) ----
        const int e0   = tile << 4;
        const int row  = lane & 15;
        const int kofs = (lane < 16) ? 0 : 2;
        const float* arow = rel + (size_t)(e0 + row) * D_REL + kofs;

        v8f acc = {};
        #pragma unroll
        for (int c = 0; c < 16; ++c) {
            const int k0 = c * 4;
            v2f a;  a.x  = arow[k0];        a.y  = arow[k0 + 1];
            v2f bb; bb.x = Wr[k0 + kofs];   bb.y = Wr[k0 + kofs + 1];
            // (neg_a, A, neg_b, B, c_mod, C, reuse_a, reuse_b)
            acc = __builtin_amdgcn_wmma_f32_16x16x4_f32(
                false, a, false, bb, (short)0, acc, false, false);
        }

        // D layout: VGPR i, lanes 0-15 -> M=i ; lanes 16-31 -> M=i+8.
        // Select component (lane&7), then shuffle so lane L (0..15) holds
        // the score of edge e0+L.
        const int r = lane & 7;
        float x = acc[0];
        x = (r == 1) ? acc[1] : x;
        x = (r == 2) ? acc[2] : x;
        x = (r == 3) ? acc[3] : x;
        x = (r == 4) ? acc[4] : x;
        x = (r == 5) ? acc[5] : x;
        x = (r == 6) ? acc[6] : x;
        x = (r == 7) ? acc[7] : x;
        const int src = (lane < 8) ? lane : (lane + 8); // lanes>=16: don't-care
        float y = __shfl(x, src, 32);

        if (lane < 16) {
            const int e   = e0 + lane;
            const float raw = y + qs;
            const bool  m   = (mask[e] != 0) && (source[e] == basin);
            scores[e]  = m ? raw * conf[e] * weight[e] : 0.0f;
            tgt_out[e] = m ? (float)target[e] : -1.0f;
        }
    } else {
        // -------- scalar tail (partial tile) --------
        const int e = (tile << 4) + lane;
        if (lane < 16 && e < E) {
            const float* arow = rel + (size_t)e * D_REL;
            float dot = 0.0f;
            #pragma unroll
            for (int k = 0; k < D_REL; ++k)
                dot = fmaf(arow[k], Wr[k], dot);
            const float raw = dot + qs;
            const bool  m   = (mask[e] != 0) && (source[e] == basin);
            scores[e]  = m ? raw * conf[e] * weight[e] : 0.0f;
            tgt_out[e] = m ? (float)target[e] : -1.0f;
        }
    }
}

// ---------------- host-side launch ----------------
extern "C" void kernel_launch(void* const* d_in, const int* in_sizes, int n_in,
                              void* d_out, int out_size, void* d_ws, size_t ws_size,
                              hipStream_t stream) {
    const float* query  = (const float*)d_in[0];         // (512,)
    const float* weight = (const float*)d_in[1];         // (E,)
    const float* conf   = (const float*)d_in[2];         // (E,)
    const float* rel    = (const float*)d_in[3];         // (E, 64)
    const float* W      = (const float*)d_in[4];         // (576,)
    const float* b      = (const float*)d_in[5];         // (1,)
    const int*   source = (const int*)d_in[6];           // (E,)
    const int*   target = (const int*)d_in[7];           // (E,)
    const unsigned char* mask = (const unsigned char*)d_in[8]; // (E,) bool
    const int*   basin  = (const int*)d_in[9];           // scalar

    const int E       = in_sizes[1];
    const int d_model = in_sizes[0];

    float* qsb     = (float*)d_ws;
    float* scores  = (float*)d_out;        // outputs concatenated: scores | targets
    float* tgt_out = (float*)d_out + E;

    qscore_kernel<<<1, 512, 0, stream>>>(query, W, b, qsb, d_model);

    const int numTiles = (E + 15) >> 4;    // 16 edges per wave32 tile
    const int blocks   = (numTiles + 7) / 8;  // 8 waves per 256-thread block
    transition_kernel<<<blocks, 256, 0, stream>>>(
        rel, W + d_model, weight, conf, source, target, mask, qsb, basin,
        scores, tgt_out, E);
}